// InfoNCELoss_48335561949654
// MI455X (gfx1250) — compile-verified
//
#include <hip/hip_runtime.h>

// Problem constants (match reference generator)
#define P_     8
#define NP1    5        // NPAST + 1
#define N_     128
#define B_     128
#define D_     512
#define CHUNK  8        // negative rows per chunk (one per wave)
#define NCHUNK (N_ / CHUNK)   // 16

typedef int v4i_t __attribute__((ext_vector_type(4)));

// ---- CDNA5 async global->LDS copy (16B per lane), guarded ----
__device__ __forceinline__ void async_cp16(const float* g, float* l) {
#if defined(__has_builtin)
#if __has_builtin(__builtin_amdgcn_global_load_async_to_lds_b128)
    // Confirmed signature on this toolchain: (v4i* gsrc, v4i* ldst, Ii offset, Ii cpol)
    __builtin_amdgcn_global_load_async_to_lds_b128(
        (v4i_t*)g, (v4i_t*)l, /*offset=*/0, /*cpol=*/0);
    return;
#endif
#endif
    *(float4*)l = *(const float4*)g;   // synchronous fallback
}

template <int CNT>
__device__ __forceinline__ void wait_async() {
#if defined(__has_builtin)
#if __has_builtin(__builtin_amdgcn_s_wait_asynccnt)
    __builtin_amdgcn_s_wait_asynccnt(CNT);
    return;
#endif
#endif
    asm volatile("s_wait_asynccnt %0" ::"i"(CNT) : "memory");
}

// ---------------------------------------------------------------
// Kernel 1: one block per (p,b). Computes per_step[p,b] and flag[p,b].
// ---------------------------------------------------------------
__global__ __launch_bounds__(256) void infonce_main(
    const float* __restrict__ pred,    // [P,B,D]
    const float* __restrict__ pos,     // [P,B,D]
    const float* __restrict__ past,    // [NP1,B,D]
    const int*   __restrict__ tIdx,    // [P,B,N]
    const int*   __restrict__ nbRaw,   // [P,B,N]
    float* __restrict__ perStep,       // [P*B]
    float* __restrict__ flags)         // [P*B]
{
    __shared__ __align__(16) float s_pred[D_];
    __shared__ __align__(16) float s_rows[2][CHUNK][D_];   // 32 KB double buffer
    __shared__ int   s_t[N_];
    __shared__ int   s_nb[N_];
    __shared__ float s_red[8];
    __shared__ float s_fkpos;
    __shared__ float s_wsum[8];
    __shared__ float s_wflag[8];

    const int pb   = blockIdx.x;           // p*B + b
    const int b    = pb & (B_ - 1);
    const int tid  = threadIdx.x;
    const int wave = tid >> 5;
    const int lane = tid & 31;

    const float* predRow = pred + (size_t)pb * D_;
    const float* posRow  = pos  + (size_t)pb * D_;

    // --- gather indices for all 128 negatives of this (p,b) ---
    if (tid < N_) {
        int t = tIdx[(size_t)pb * N_ + tid];
        int r = nbRaw[(size_t)pb * N_ + tid];
        s_t[tid]  = t;
        s_nb[tid] = r + ((r >= b) ? 1 : 0);   // skip-self batch index
    }

    // --- stage prediction row to LDS; fk_pos partials on the way ---
    float a0 = predRow[tid];
    float a1 = predRow[tid + 256];
    s_pred[tid]       = a0;
    s_pred[tid + 256] = a1;
    float pp = __expf(a0 * posRow[tid]) + __expf(a1 * posRow[tid + 256]);
#pragma unroll
    for (int off = 16; off >= 1; off >>= 1) pp += __shfl_xor(pp, off);
    if (lane == 0) s_red[wave] = pp;
    __syncthreads();                       // s_t / s_nb / s_pred / s_red ready

    if (tid == 0) {
        float s = 0.f;
#pragma unroll
        for (int w = 0; w < 8; ++w) s += s_red[w];
        s_fkpos = s;                       // visible after next barrier
    }

    // cache this lane's slice of the prediction row in registers:
    // d = lane*4 + j*128 (contiguous float4 per 128-float block)
    float4 pv[4];
#pragma unroll
    for (int j = 0; j < 4; ++j)
        pv[j] = *(const float4*)&s_pred[lane * 4 + j * 128];

    // --- async prefetch of one chunk (8 rows * 2KB) into LDS buffer ---
    auto prefetch = [&](int c, int buf) {
        const int base = c * CHUNK;
#pragma unroll
        for (int i = 0; i < 4; ++i) {
            int idx = i * 256 + tid;       // 0..1023 float4 slots
            int r   = idx >> 7;            // row within chunk
            int f   = idx & 127;           // float4 within row
            int t   = s_t[base + r];
            int nb  = s_nb[base + r];
            const float* src = (t < NP1)
                ? past + (size_t)(t * B_ + nb) * D_
                : pos  + (size_t)((t - NP1) * B_ + nb) * D_;
            async_cp16(src + f * 4, &s_rows[buf][r][f * 4]);
        }
    };

    prefetch(0, 0);

    float wsum = 0.f;                      // per-wave sum of fk_neg
    int   wok  = 1;                        // per-wave AND of (fk_pos > fk_neg)

    for (int c = 0; c < NCHUNK; ++c) {
        const int buf = c & 1;
        if (c + 1 < NCHUNK) {
            prefetch(c + 1, buf ^ 1);      // overlap scattered L2 gather with exp stream
            wait_async<4>();               // retire chunk c (async loads complete in order)
        } else {
            wait_async<0>();
        }
        __syncthreads();                   // chunk c fully in LDS for everyone

        // wave 'wave' reduces negative row (c*8 + wave); ds_load_b128 reads
        const float* row = &s_rows[buf][wave][0];
        float acc = 0.f;
#pragma unroll
        for (int j = 0; j < 4; ++j) {
            float4 rv = *(const float4*)&row[lane * 4 + j * 128];
            acc += __expf(pv[j].x * rv.x);
            acc += __expf(pv[j].y * rv.y);
            acc += __expf(pv[j].z * rv.z);
            acc += __expf(pv[j].w * rv.w);
        }
#pragma unroll
        for (int off = 16; off >= 1; off >>= 1) acc += __shfl_xor(acc, off);
        if (lane == 0) {
            wsum += acc;
            wok  &= (s_fkpos > acc) ? 1 : 0;
        }
        __syncthreads();                   // all reads of buf done before reuse
    }

    if (lane == 0) { s_wsum[wave] = wsum; s_wflag[wave] = (float)wok; }
    __syncthreads();

    if (tid == 0) {
        float sn = 0.f, fl = 1.f;
#pragma unroll
        for (int w = 0; w < 8; ++w) { sn += s_wsum[w]; fl *= s_wflag[w]; }
        float fkp = s_fkpos;
        perStep[pb] = __logf(fkp / (fkp + sn)) / (float)(N_ + 1);
        flags[pb]   = fl;
    }
}

// ---------------------------------------------------------------
// Kernel 2: reduce 1024 per-step values -> (loss, correct)
// ---------------------------------------------------------------
__global__ __launch_bounds__(256) void infonce_reduce(
    const float* __restrict__ ws, float* __restrict__ out)
{
    __shared__ float sl[8], sc[8];
    const int tid  = threadIdx.x;
    const int wave = tid >> 5;
    const int lane = tid & 31;

    float ls = 0.f, cs = 0.f;
#pragma unroll
    for (int i = 0; i < 4; ++i) {
        int idx = tid + i * 256;
        ls += ws[idx];                 // per_step
        cs += ws[P_ * B_ + idx];       // flags
    }
#pragma unroll
    for (int off = 16; off >= 1; off >>= 1) {
        ls += __shfl_xor(ls, off);
        cs += __shfl_xor(cs, off);
    }
    if (lane == 0) { sl[wave] = ls; sc[wave] = cs; }
    __syncthreads();
    if (tid == 0) {
        float L = 0.f, C = 0.f;
#pragma unroll
        for (int w = 0; w < 8; ++w) { L += sl[w]; C += sc[w]; }
        out[0] = -L / (float)(P_ * B_);   // loss
        out[1] =  C / (float)P_;          // correct (mean over P of batch counts)
    }
}

// ---------------------------------------------------------------
extern "C" void kernel_launch(void* const* d_in, const int* in_sizes, int n_in,
                              void* d_out, int out_size, void* d_ws, size_t ws_size,
                              hipStream_t stream) {
    const float* pred  = (const float*)d_in[0];   // latent_predictions [P,B,D]
    const float* pos   = (const float*)d_in[1];   // positive_samples   [P,B,D]
    const float* past  = (const float*)d_in[2];   // past_latents       [NP1,B,D]
    const int*   tIdx  = (const int*)d_in[3];     // time_idx           [P,B,N]
    const int*   nbRaw = (const int*)d_in[4];     // neg_b_raw          [P,B,N]
    (void)in_sizes; (void)n_in; (void)out_size; (void)ws_size;

    float* ws  = (float*)d_ws;                    // [0,1024): per_step, [1024,2048): flags
    float* out = (float*)d_out;                   // {loss, correct}

    infonce_main<<<P_ * B_, 256, 0, stream>>>(pred, pos, past, tIdx, nbRaw,
                                              ws, ws + P_ * B_);
    infonce_reduce<<<1, 256, 0, stream>>>(ws, out);
}